// MaskAndAtt_78975858639293
// MI455X (gfx1250) — compile-verified
//
#include <hip/hip_runtime.h>
#include <hip/hip_bf16.h>

// Problem dims (fixed by the reference)
#define BB 32
#define TT 2048
#define HH 512
#define DD 1024    // 2*HH
#define SLICES 16  // T-slices for the two streaming reductions

typedef float v2f __attribute__((ext_vector_type(2)));
typedef float v8f __attribute__((ext_vector_type(8)));

// ---------------------------------------------------------------------------
// Pass 1a: hsum partials.  part[s][b][d] = sum over one T-slice of
//          h[b,t,d] * mask[b,t,d & (H-1)].  float4 per thread -> b128 loads;
//          one block covers all of D (256 threads * 4), grid = B*SLICES.
// ---------------------------------------------------------------------------
__global__ void hsum_part_kernel(const float* __restrict__ h,
                                 const float* __restrict__ mask,
                                 float* __restrict__ part) {
    const int s   = blockIdx.x % SLICES;
    const int b   = blockIdx.x / SLICES;
    const int d4  = threadIdx.x * 4;        // 0..1020
    const int md4 = d4 & (HH - 1);          // 16B-aligned mask offset

    float4 acc = {0.f, 0.f, 0.f, 0.f};
    const int t0 = s * (TT / SLICES);
    for (int t = t0; t < t0 + (TT / SLICES); ++t) {
        const float4 hv = *(const float4*)(h    + (size_t)(b * TT + t) * DD + d4);
        const float4 mv = *(const float4*)(mask + (size_t)(b * TT + t) * HH + md4);
        acc.x += hv.x * mv.x;
        acc.y += hv.y * mv.y;
        acc.z += hv.z * mv.z;
        acc.w += hv.w * mv.w;
    }
    *(float4*)(part + ((size_t)s * BB + b) * DD + d4) = acc;
}

// ---------------------------------------------------------------------------
// Sum SLICES partial planes of n elements each into out (float4 lanes).
// ---------------------------------------------------------------------------
__global__ void reduce_slices_kernel(const float* __restrict__ part,
                                     float* __restrict__ out, int n) {
    const int i4 = (blockIdx.x * blockDim.x + threadIdx.x) * 4;
    if (i4 < n) {
        float4 a = {0.f, 0.f, 0.f, 0.f};
        for (int s = 0; s < SLICES; ++s) {
            const float4 p = *(const float4*)(part + (size_t)s * n + i4);
            a.x += p.x; a.y += p.y; a.z += p.z; a.w += p.w;
        }
        *(float4*)(out + i4) = a;
    }
}

// ---------------------------------------------------------------------------
// Pass 2: beta[b,t] = lstm[b,t,:] . hsum[b,:] via V_WMMA_F32_16X16X4_F32.
// One wave owns 16 consecutive t-rows.  A-layout (ISA 32-bit A 16x4):
//   lane m = lane&15 holds row M=m; half = lane>>4 selects K-pair;
//   VGPR0 = K(2*half), VGPR1 = K(2*half+1).
// B is broadcast so every column holds hsum -> all 16 columns of D equal
// beta for that row; extraction uses only the documented C/D layout:
//   lanes 0-15: VGPR r = (M=r, N=lane); lanes 16-31: VGPR r = (M=8+r).
// EXEC is all ones through the whole WMMA loop (wave32, no divergence).
// Codegen verified: 16x v_wmma_f32_16x16x4_f32 pipelined against b64 loads.
// ---------------------------------------------------------------------------
__global__ void beta_wmma_kernel(const float* __restrict__ lstm,
                                 const float* __restrict__ hsum,
                                 float* __restrict__ beta) {
    const int tile = blockIdx.x % (TT / 128);
    const int b    = blockIdx.x / (TT / 128);
    const int wave = threadIdx.x >> 5;   // 8 waves / block
    const int lane = threadIdx.x & 31;
    const int m    = lane & 15;
    const int half = lane >> 4;          // 0 or 1 -> K-pair select
    const int t0   = tile * 128 + wave * 16;

    const float* __restrict__ lrow = lstm + (size_t)(b * TT + t0 + m) * DD;
    const float* __restrict__ hs   = hsum + (size_t)b * DD;

    v8f c = {0.f, 0.f, 0.f, 0.f, 0.f, 0.f, 0.f, 0.f};
    for (int k = 0; k < DD; k += 4) {
        // 8-byte aligned: row base is 4KB aligned, k%4==0, 2*half even.
        const v2f a  = *(const v2f*)(lrow + k + 2 * half);
        const v2f bv = *(const v2f*)(hs   + k + 2 * half);
        c = __builtin_amdgcn_wmma_f32_16x16x4_f32(
                /*neg_a=*/false, a, /*neg_b=*/false, bv,
                /*c_mod=*/(short)0, c, /*reuse_a=*/false, /*reuse_b=*/false);
    }
    // lane 0 carries rows t0..t0+7 in c[0..7]; lane 16 carries t0+8..t0+15.
    if (m == 0) {
        float* out = beta + (size_t)b * TT + t0 + 8 * half;
#pragma unroll
        for (int r = 0; r < 8; ++r) out[r] = c[r];
    }
}

// ---------------------------------------------------------------------------
// Pass 3: softmax over T per batch.  One block per b; 8 values/thread kept
// in registers; two LDS tree reductions (max, sum).
// ---------------------------------------------------------------------------
__global__ void softmax_kernel(const float* __restrict__ beta,
                               float* __restrict__ alpha) {
    __shared__ float red[256];
    const int b   = blockIdx.x;
    const int tid = threadIdx.x;
    const int PT  = TT / 256;  // 8

    float v[PT];
    float mx = -3.402823466e+38f;
#pragma unroll
    for (int i = 0; i < PT; ++i) {
        v[i] = beta[(size_t)b * TT + tid + i * 256];
        mx = fmaxf(mx, v[i]);
    }
    red[tid] = mx;
    __syncthreads();
    for (int off = 128; off > 0; off >>= 1) {
        if (tid < off) red[tid] = fmaxf(red[tid], red[tid + off]);
        __syncthreads();
    }
    mx = red[0];
    __syncthreads();

    float sum = 0.0f;
#pragma unroll
    for (int i = 0; i < PT; ++i) {
        v[i] = __expf(v[i] - mx);
        sum += v[i];
    }
    red[tid] = sum;
    __syncthreads();
    for (int off = 128; off > 0; off >>= 1) {
        if (tid < off) red[tid] += red[tid + off];
        __syncthreads();
    }
    const float inv = 1.0f / red[0];
#pragma unroll
    for (int i = 0; i < PT; ++i)
        alpha[(size_t)b * TT + tid + i * 256] = v[i] * inv;
}

// ---------------------------------------------------------------------------
// Pass 4a: r partials.  part[s][b][d] = sum over one T-slice of
//          alpha[b,t] * lstm[b,t,d].  float4 lstm loads; alpha is a uniform
//          scalar per t (scalar-load + broadcast).
// ---------------------------------------------------------------------------
__global__ void r_part_kernel(const float* __restrict__ lstm,
                              const float* __restrict__ alpha,
                              float* __restrict__ part) {
    const int s  = blockIdx.x % SLICES;
    const int b  = blockIdx.x / SLICES;
    const int d4 = threadIdx.x * 4;

    float4 acc = {0.f, 0.f, 0.f, 0.f};
    const int t0 = s * (TT / SLICES);
    for (int t = t0; t < t0 + (TT / SLICES); ++t) {
        const float  av = alpha[(size_t)b * TT + t];
        const float4 lv = *(const float4*)(lstm + (size_t)(b * TT + t) * DD + d4);
        acc.x += av * lv.x;
        acc.y += av * lv.y;
        acc.z += av * lv.z;
        acc.w += av * lv.w;
    }
    *(float4*)(part + ((size_t)s * BB + b) * DD + d4) = acc;
}

// ---------------------------------------------------------------------------
// Launch: inputs are h [B,T,D] f32, lstm [B,T,D] f32, mask [B,T,H] f32.
// d_out = r (B*D floats) then alpha (B*T floats).
// ws layout: [ part SLICES*B*D | hsum B*D | beta B*T ]  (~2.4 MB); the
// partials buffer is reused by pass 4 after pass 1 is consumed.
// ---------------------------------------------------------------------------
extern "C" void kernel_launch(void* const* d_in, const int* in_sizes, int n_in,
                              void* d_out, int out_size, void* d_ws, size_t ws_size,
                              hipStream_t stream) {
    const float* h    = (const float*)d_in[0];
    const float* lstm = (const float*)d_in[1];
    const float* mask = (const float*)d_in[2];

    float* out_r     = (float*)d_out;                       // B*D
    float* out_alpha = (float*)d_out + (size_t)BB * DD;     // B*T

    float* ws_part = (float*)d_ws;                          // SLICES*B*D (reused)
    float* ws_hsum = ws_part + (size_t)SLICES * BB * DD;    // B*D
    float* ws_beta = ws_hsum + (size_t)BB * DD;             // B*T

    const int nBD = BB * DD;
    const int reduceBlocks = (nBD / 4 + 255) / 256;

    hsum_part_kernel<<<BB * SLICES, 256, 0, stream>>>(h, mask, ws_part);
    reduce_slices_kernel<<<reduceBlocks, 256, 0, stream>>>(ws_part, ws_hsum, nBD);
    beta_wmma_kernel<<<BB * (TT / 128), 256, 0, stream>>>(lstm, ws_hsum, ws_beta);
    softmax_kernel<<<BB, 256, 0, stream>>>(ws_beta, out_alpha);
    r_part_kernel<<<BB * SLICES, 256, 0, stream>>>(lstm, out_alpha, ws_part);
    reduce_slices_kernel<<<reduceBlocks, 256, 0, stream>>>(ws_part, out_r, nBD);
}